// LSTM_Univariate_79018808312412
// MI455X (gfx1250) — compile-verified
//
#include <hip/hip_runtime.h>
#include <stdint.h>

#define T_STEPS       32768
#define F_FEAT        512
#define BLOCK_THREADS 128
#define CHUNK         32                    // time steps per double-buffered chunk
#define ROWS_PER_WAVE (CHUNK / 4)           // 8 rows per wave's TDM tile
#define NCHUNKS       (T_STEPS / CHUNK)

typedef unsigned int       u32;
typedef unsigned long long u64;
typedef u32 v4u __attribute__((ext_vector_type(4)));
typedef int v4i __attribute__((ext_vector_type(4)));
typedef int v8i __attribute__((ext_vector_type(8)));

// ---- feature detection -----------------------------------------------------
#if defined(__has_builtin)
#if __has_builtin(__builtin_amdgcn_tensor_load_to_lds) && \
    __has_builtin(__builtin_amdgcn_tensor_store_from_lds)
#define HAVE_TDM 1
#endif
#endif
#ifndef HAVE_TDM
#define HAVE_TDM 0
#endif

#if __has_include(<hip/amd_detail/amd_gfx1250_TDM.h>)
#define TDM_6ARG 1
#else
#define TDM_6ARG 0
#endif

// ---- fast math on gfx1250 --------------------------------------------------
__device__ __forceinline__ float fast_exp2(float x) {
#if defined(__has_builtin) && __has_builtin(__builtin_amdgcn_exp2f)
  return __builtin_amdgcn_exp2f(x);
#else
  return exp2f(x);
#endif
}
__device__ __forceinline__ float fast_rcp(float x) {
#if defined(__has_builtin) && __has_builtin(__builtin_amdgcn_rcpf)
  return __builtin_amdgcn_rcpf(x);
#else
  return 1.0f / x;
#endif
}
// sigmoid with the -log2e scale pre-folded into the gate pre-activation:
// caller passes g' = -log2e * g, we return 1/(1+2^g')
__device__ __forceinline__ float fsig_pre(float gp) {
  return fast_rcp(1.0f + fast_exp2(gp));
}
__device__ __forceinline__ float ftanh(float x) {
#if defined(__has_builtin) && __has_builtin(__builtin_amdgcn_tanhf)
  return __builtin_amdgcn_tanhf(x);   // v_tanh_f32 (gfx1250 trans op)
#else
  return 2.0f * fast_rcp(1.0f + fast_exp2(x * -2.885390081777927f)) - 1.0f;
#endif
}

// s_wait_tensorcnt takes an immediate -> template parameter, constant at call.
template <int N>
__device__ __forceinline__ void wait_tensorcnt() {
#if defined(__has_builtin) && __has_builtin(__builtin_amdgcn_s_wait_tensorcnt)
  __builtin_amdgcn_s_wait_tensorcnt(N);
#else
  asm volatile("s_wait_tensorcnt %0" :: "i"(N) : "memory");
#endif
}

// ---- TDM descriptors: rows x 128 f32 tile, row stride F_FEAT ---------------
#if HAVE_TDM
__device__ __forceinline__ void tdm_desc(u64 gaddr, u32 lds_addr, u32 rows,
                                         v4u& g0, v8i& g1) {
  g0.x = 1u;                                         // count=1 (user mode)
  g0.y = lds_addr;                                   // LDS byte address
  g0.z = (u32)(gaddr & 0xFFFFFFFFull);               // global_addr[31:0]
  g0.w = ((u32)(gaddr >> 32) & 0x01FFFFFFu)          // global_addr[56:32]
         | 0x80000000u;                              // type=2 ("image")
  const int dim0 = BLOCK_THREADS;                    // 128 elements per row
  g1 = (v8i){
      (int)(2u << 16),                // data_size=2 (4B); wg_mask=0; no flags
      (int)((u32)dim0 << 16),         // tensor_dim0[15:0] @ bits 63:48
      (int)(rows << 16),              // tensor_dim1[15:0] @ bits 95:80
      (int)((u32)dim0 << 16),         // tile_dim0 @ bits 127:112
      (int)rows,                      // tile_dim1 @ bits 143:128 (tile_dim2=0)
      (int)F_FEAT,                    // tensor_dim0_stride[31:0] @ bits 191:160
      0, 0
  };
}
__device__ __forceinline__ void tdm_load_tile(u64 gaddr, u32 lds_addr, u32 rows) {
  v4u g0; v8i g1;
  tdm_desc(gaddr, lds_addr, rows, g0, g1);
  v4i g2 = {0, 0, 0, 0}, g3 = {0, 0, 0, 0};
#if TDM_6ARG
  v8i gx = {0, 0, 0, 0, 0, 0, 0, 0};
  __builtin_amdgcn_tensor_load_to_lds(g0, g1, g2, g3, gx, 0);
#else
  __builtin_amdgcn_tensor_load_to_lds(g0, g1, g2, g3, 0);
#endif
}
__device__ __forceinline__ void tdm_store_tile(u64 gaddr, u32 lds_addr, u32 rows) {
  v4u g0; v8i g1;
  tdm_desc(gaddr, lds_addr, rows, g0, g1);
  v4i g2 = {0, 0, 0, 0}, g3 = {0, 0, 0, 0};
#if TDM_6ARG
  v8i gx = {0, 0, 0, 0, 0, 0, 0, 0};
  __builtin_amdgcn_tensor_store_from_lds(g0, g1, g2, g3, gx, 0);
#else
  __builtin_amdgcn_tensor_store_from_lds(g0, g1, g2, g3, 0);
#endif
}
#endif

// ---- kernel ----------------------------------------------------------------
__global__ __launch_bounds__(BLOCK_THREADS)
void lstm_univariate_kernel(const float* __restrict__ x,
                            const float* __restrict__ w_ih,
                            const float* __restrict__ w_hh,
                            const float* __restrict__ b_ih,
                            const float* __restrict__ b_hh,
                            const float* __restrict__ h0,
                            const float* __restrict__ c0,
                            float* __restrict__ out) {
  __shared__ float xs[2][CHUNK * BLOCK_THREADS];     // 2 x 16 KB input staging
  __shared__ float ys[2][CHUNK * BLOCK_THREADS];     // 2 x 16 KB output staging

  const int tid   = threadIdx.x;
  const int fglob = blockIdx.x * BLOCK_THREADS + tid;
  const u32 wave  = (u32)__builtin_amdgcn_readfirstlane((int)((u32)threadIdx.x >> 5));

  // Per-feature parameters (vectorized 16B loads; [F,4] is contiguous).
  const float4 wi = ((const float4*)w_ih)[fglob];
  const float4 wh = ((const float4*)w_hh)[fglob];
  const float4 bi = ((const float4*)b_ih)[fglob];
  const float4 bh = ((const float4*)b_hh)[fglob];
  // Sigmoid gates (i, f, o): pre-fold -log2e so the per-step chain is
  // fma -> fma -> v_exp -> add -> v_rcp with no extra dependent v_mul.
  const float NL2E = -1.4426950408889634f;
  const float wix = wi.x * NL2E, whx = wh.x * NL2E, bx = (bi.x + bh.x) * NL2E;
  const float wif = wi.y * NL2E, whf = wh.y * NL2E, bf = (bi.y + bh.y) * NL2E;
  const float wio = wi.w * NL2E, who = wh.w * NL2E, bo = (bi.w + bh.w) * NL2E;
  // Tanh gate (g): raw scale for v_tanh_f32.
  const float wig = wi.z, whg = wh.z, bg = bi.z + bh.z;

  float h = h0[fglob];
  float c = c0[fglob];

  // Tile bases. Each wave DMAs its 8-row quarter of each chunk tile.
  const u64 xbase = (u64)(uintptr_t)x   + (u64)blockIdx.x * BLOCK_THREADS * 4ull;
  const u64 obase = (u64)(uintptr_t)out + (u64)blockIdx.x * BLOCK_THREADS * 4ull;
  const u32 ldsx0 = (u32)(uintptr_t)&xs[0][0];
  const u32 ldsx1 = (u32)(uintptr_t)&xs[1][0];
  const u32 ldsy0 = (u32)(uintptr_t)&ys[0][0];
  const u32 ldsy1 = (u32)(uintptr_t)&ys[1][0];
  const u32 wave_lds_off = wave * (ROWS_PER_WAVE * BLOCK_THREADS * 4);
  const u64 wave_g_off   = (u64)(wave * ROWS_PER_WAVE) * F_FEAT * 4ull;

#if HAVE_TDM
  // Prologue: DMA chunk 0 into xs[0].
  tdm_load_tile(xbase + wave_g_off, ldsx0 + wave_lds_off, ROWS_PER_WAVE);
  wait_tensorcnt<0>();
#else
  for (int r = 0; r < CHUNK; ++r)
    xs[0][r * BLOCK_THREADS + tid] = x[(u64)r * F_FEAT + fglob];
#endif
  __syncthreads();

  for (int k = 0; k < NCHUNKS; ++k) {
    const int kb = k & 1;
#if HAVE_TDM
    if (k + 1 < NCHUNKS) {
      const u32 ldsxn = (kb ? ldsx0 : ldsx1);
      tdm_load_tile(xbase + (u64)(k + 1) * CHUNK * F_FEAT * 4ull + wave_g_off,
                    ldsxn + wave_lds_off, ROWS_PER_WAVE);
    }
#endif
    const float* xcol = &xs[kb][tid];
    float*       ycol = &ys[kb][tid];
#pragma unroll 4
    for (int t = 0; t < CHUNK; ++t) {
      const float xv = xcol[t * BLOCK_THREADS];      // ds_load_b32
      const float gi = fmaf(xv, wix, fmaf(h, whx, bx));
      const float gf = fmaf(xv, wif, fmaf(h, whf, bf));
      const float gg = fmaf(xv, wig, fmaf(h, whg, bg));
      const float go = fmaf(xv, wio, fmaf(h, who, bo));
      const float ig = fsig_pre(gi);
      const float fg = fsig_pre(gf);
      const float og = fsig_pre(go);
      const float gv = ftanh(gg);
      c = fmaf(fg, c, ig * gv);
      h = og * ftanh(c);
      ycol[t * BLOCK_THREADS] = 2.0f * h;            // ds_store_b32
    }
    __syncthreads();  // ys[kb] fully written; xs[kb] fully consumed
#if HAVE_TDM
    tdm_store_tile(obase + (u64)k * CHUNK * F_FEAT * 4ull + wave_g_off,
                   (kb ? ldsy1 : ldsy0) + wave_lds_off, ROWS_PER_WAVE);
    // In-order TENSORcnt: <=1 retires S(k-1) and L(k+1), leaving only S(k).
    wait_tensorcnt<1>();
#else
    for (int r = 0; r < CHUNK; ++r)
      out[((u64)k * CHUNK + r) * F_FEAT + fglob] = ys[kb][r * BLOCK_THREADS + tid];
    if (k + 1 < NCHUNKS) {
      const int kn = kb ^ 1;
      for (int r = 0; r < CHUNK; ++r)
        xs[kn][r * BLOCK_THREADS + tid] =
            x[((u64)(k + 1) * CHUNK + r) * F_FEAT + fglob];
    }
#endif
    __syncthreads();
  }
  // s_endpgm performs an implicit wait-idle, draining the final TDM store.
}

// ---- launch ----------------------------------------------------------------
extern "C" void kernel_launch(void* const* d_in, const int* in_sizes, int n_in,
                              void* d_out, int out_size, void* d_ws, size_t ws_size,
                              hipStream_t stream) {
  (void)in_sizes; (void)n_in; (void)out_size; (void)d_ws; (void)ws_size;
  const float* x    = (const float*)d_in[0];
  const float* w_ih = (const float*)d_in[1];
  const float* w_hh = (const float*)d_in[2];
  const float* b_ih = (const float*)d_in[3];
  const float* b_hh = (const float*)d_in[4];
  const float* h0   = (const float*)d_in[5];
  const float* c0   = (const float*)d_in[6];
  float* out = (float*)d_out;

  dim3 grid(F_FEAT / BLOCK_THREADS);   // 4 blocks, one 128-feature slab each
  dim3 block(BLOCK_THREADS);
  lstm_univariate_kernel<<<grid, block, 0, stream>>>(x, w_ih, w_hh, b_ih, b_hh,
                                                     h0, c0, out);
}